// GwcVolume_15487652069510
// MI455X (gfx1250) — compile-verified
//
#include <hip/hip_runtime.h>

typedef float v2f __attribute__((ext_vector_type(2)));
typedef float v8f __attribute__((ext_vector_type(8)));

namespace {
constexpr int G  = 40;    // correlation groups
constexpr int Cg = 8;     // channels per group (F=320 = G*Cg)
constexpr int Hh = 128;
constexpr int Ww = 256;
constexpr int Dd = 48;    // disparity bins (bins == arange(48))
constexpr int HW = Hh * Ww;
constexpr int PAD = 64;   // LDS pad: scatter trash slot + unguarded rl reads
}

// One workgroup per (g, h). Phase 1: build cost[d][w] = (1/8) sum_c L[w,c]*R[w-d,c]
// via f32 WMMA banded-Toeplitz tiles, scattering the diagonal band into LDS.
// Phase 2: stream lr/rl out of the LDS cost plane with coalesced float4 stores.
__global__ __launch_bounds__(128)
void gwc_volume_kernel(const float* __restrict__ left,
                       const float* __restrict__ right,
                       float* __restrict__ out_lr,
                       float* __restrict__ out_rl) {
  __shared__ float cost[Dd * Ww + PAD];   // 48 KiB + pad

  const int g   = blockIdx.x / Hh;
  const int h   = blockIdx.x % Hh;
  const int tid = threadIdx.x;
  const int wid = tid >> 5;         // wave id (wave32)
  const int ln  = tid & 31;
  const int hf  = ln >> 4;          // half-wave (selects K group / M group)
  const int n16 = ln & 15;

  // (g,h) slab base: element (c, w) lives at base + c*HW + w
  const float* Lp = left  + (size_t)g * Cg * HW + (size_t)h * Ww;
  const float* Rp = right + (size_t)g * Cg * HW + (size_t)h * Ww;

  // f32 WMMA A/B operand layout: VGPR0 holds K = {0 | 2}, VGPR1 holds K = {1 | 3}
  // for lanes 0-15 | 16-31 respectively.
  const int c0 = hf * 2;

  // 48 tiles (3 d-tiles x 16 w-tiles), 12 per wave.
  for (int t = wid * 12; t < wid * 12 + 12; ++t) {
    const int d0 = (t >> 4) << 4;
    const int w0 = (t & 15) << 4;

    // A = L rows [w0, w0+16), K = channel
    const int rowA = w0 + n16;          // M = lane % 16
    v2f a0, a1;
    a0.x = Lp[(size_t)(c0    ) * HW + rowA];
    a0.y = Lp[(size_t)(c0 + 1) * HW + rowA];
    a1.x = Lp[(size_t)(c0 + 4) * HW + rowA];
    a1.y = Lp[(size_t)(c0 + 5) * HW + rowA];

    // B tiles = R rows [wbase, wbase+16) and [wbase+16, wbase+32), wbase = w0-d0-16
    const int wp0 = w0 - d0 - 16 + n16; // w' for P0 column n16 (may be < 0 -> masked)
    const int wp1 = w0 - d0      + n16; // w' for P1 column n16
    const int rB0 = wp0 < 0 ? 0 : wp0;  // clamp; garbage is routed to trash slot
    const int rB1 = wp1 < 0 ? 0 : wp1;

    v2f b00, b01, b10, b11;
    b00.x = Rp[(size_t)(c0    ) * HW + rB0];
    b00.y = Rp[(size_t)(c0 + 1) * HW + rB0];
    b01.x = Rp[(size_t)(c0 + 4) * HW + rB0];
    b01.y = Rp[(size_t)(c0 + 5) * HW + rB0];
    b10.x = Rp[(size_t)(c0    ) * HW + rB1];
    b10.y = Rp[(size_t)(c0 + 1) * HW + rB1];
    b11.x = Rp[(size_t)(c0 + 4) * HW + rB1];
    b11.y = Rp[(size_t)(c0 + 5) * HW + rB1];

    // P[i,n] = sum_{c=0..7} L[w0+i, c] * R[wbase+n, c]; K=8 via two chained K=4 steps.
    v8f acc0 = {}, acc1 = {};
    acc0 = __builtin_amdgcn_wmma_f32_16x16x4_f32(false, a0, false, b00, (short)0, acc0, false, false);
    acc0 = __builtin_amdgcn_wmma_f32_16x16x4_f32(false, a1, false, b01, (short)0, acc0, false, false);
    acc1 = __builtin_amdgcn_wmma_f32_16x16x4_f32(false, a0, false, b10, (short)0, acc1, false, false);
    acc1 = __builtin_amdgcn_wmma_f32_16x16x4_f32(false, a1, false, b11, (short)0, acc1, false, false);

    // Branch-free scatter of the diagonal band into the LDS cost plane.
    // D layout: element (M = v + 8*hf, N = n16) lives in acc[v].
    // For each v exactly one of acc0/acc1 is on-band:
    //   i <  n16 : acc0, d-d0 = 16 + i - n16, validity wp0 >= 0
    //   i >= n16 : acc1, d-d0 =      i - n16, validity wp1 >= 0
    // Invalid lanes (w' < 0 <=> w < d) store to the trash slot instead of branching.
#pragma unroll
    for (int v = 0; v < 8; ++v) {
      const int  i   = v + 8 * hf;
      const bool lo  = (i < n16);
      const float val = (lo ? acc0[v] : acc1[v]) * 0.125f;
      const int  dl  = i - n16 + (lo ? 16 : 0);
      const int  wp  = lo ? wp0 : wp1;
      const int  adr = (wp >= 0) ? ((d0 + dl) * Ww + w0 + i) : (Dd * Ww);
      cost[adr] = val;
    }
  }

  __syncthreads();

  // Phase 2: lr[d,w] = cost[d,w] (w >= d), rl[d,w] = cost[d,w+d] (w < W-d).
  const size_t obase = (size_t)g * Dd * HW + (size_t)h * Ww;
  float* lrp = out_lr + obase;
  float* rlp = out_rl + obase;

  for (int q = tid; q < (Dd * Ww) / 4; q += 128) {
    const int d = q >> 6;             // 64 float4 per row
    const int w = (q & 63) << 2;
    const float* cr = &cost[d * Ww];

    const float4 cv = *reinterpret_cast<const float4*>(cr + w); // masked below
    float4 lv;
    lv.x = (w + 0 >= d) ? cv.x : 0.0f;
    lv.y = (w + 1 >= d) ? cv.y : 0.0f;
    lv.z = (w + 2 >= d) ? cv.z : 0.0f;
    lv.w = (w + 3 >= d) ? cv.w : 0.0f;

    // Unconditional loads (pad keeps them in-bounds), then value-select.
    const int wd = w + d;
    const float r0 = cr[wd + 0];
    const float r1 = cr[wd + 1];
    const float r2 = cr[wd + 2];
    const float r3 = cr[wd + 3];
    float4 rv;
    rv.x = (wd + 0 < Ww) ? r0 : 0.0f;
    rv.y = (wd + 1 < Ww) ? r1 : 0.0f;
    rv.z = (wd + 2 < Ww) ? r2 : 0.0f;
    rv.w = (wd + 3 < Ww) ? r3 : 0.0f;

    *reinterpret_cast<float4*>(lrp + (size_t)d * HW + w) = lv;
    *reinterpret_cast<float4*>(rlp + (size_t)d * HW + w) = rv;
  }
}

extern "C" void kernel_launch(void* const* d_in, const int* in_sizes, int n_in,
                              void* d_out, int out_size, void* d_ws, size_t ws_size,
                              hipStream_t stream) {
  (void)in_sizes; (void)n_in; (void)d_ws; (void)ws_size; (void)out_size;
  const float* left  = (const float*)d_in[0];
  const float* right = (const float*)d_in[1];
  // d_in[2] = bins = arange(48): disparity value == bin index (hardcoded in kernel).
  float* out_lr = (float*)d_out;
  float* out_rl = out_lr + (size_t)G * Dd * HW;

  dim3 grid(G * Hh);   // one workgroup per (group, row) = 5120 blocks
  dim3 block(128);     // 4 wave32 waves
  hipLaunchKernelGGL(gwc_volume_kernel, grid, block, 0, stream,
                     left, right, out_lr, out_rl);
}